// PointConv_75754633167313
// MI455X (gfx1250) — compile-verified
//
#include <hip/hip_runtime.h>

// ---------------------------------------------------------------------------
// PointConv (ball-query + grouped 1x1 conv) for MI455X / gfx1250, wave32.
//
//   B=4, C_in=64, C_out=128, N=4096, S=9, radius=0.2
//   y[b,o,n] = bias[o] + sum_{c,s} x[b,c,idx[b,n,s]] * W[o,c,s]
// => per-batch GEMM:  [128 x 576] (W) * [576 x 4096] (gathered x)
//
// Kernel 1: ball query, one wave32 per point, __ballot extracts the first 9
//           in-radius indices in ascending order (lane order == index order).
// Kernel 2: fused gather -> LDS -> V_WMMA_F32_16X16X4_F32 GEMM, fp32 exact.
//           LDS B-panel is pair-interleaved so every fragment load is one
//           aligned b64 access with an immediate offset.
// ---------------------------------------------------------------------------

typedef __attribute__((ext_vector_type(2))) float v2f;
typedef __attribute__((ext_vector_type(8))) float v8f;

constexpr int   kB     = 4;
constexpr int   kCin   = 64;
constexpr int   kCout  = 128;
constexpr int   kN     = 4096;
constexpr int   kS     = 9;
constexpr int   kK     = kCin * kS;     // 576
constexpr int   kTileN = 16;            // points per workgroup
constexpr float kR2    = 0.2f * 0.2f;

// ---------------------------------------------------------------------------
// Kernel 1: ball query. blockDim=256 (8 waves), one wave per point.
// ---------------------------------------------------------------------------
__global__ __launch_bounds__(256) void ballquery_kernel(
    const float* __restrict__ pcs,   // [B][3][N]
    int* __restrict__ idx)           // [B][N][9]
{
  const int wid  = (blockIdx.x * blockDim.x + threadIdx.x) >> 5;  // global wave
  const int lane = threadIdx.x & 31;
  const int b    = wid >> 12;        // / 4096
  const int n    = wid & (kN - 1);

  const float* P = pcs + (size_t)b * 3 * kN;
  const float px = P[n];
  const float py = P[kN + n];
  const float pz = P[2 * kN + n];

  int found[kS];
  int cnt = 0;

  for (int base = 0; base < kN && cnt < kS; base += 32) {
    const int m = base + lane;
    const float dx = P[m]          - px;
    const float dy = P[kN + m]     - py;
    const float dz = P[2 * kN + m] - pz;
    const float d2 = dx * dx + dy * dy + dz * dz;
    unsigned mk = (unsigned)__ballot(d2 <= kR2);   // wave32: low 32 bits
    while (mk && cnt < kS) {
      const int bit = __ffs(mk) - 1;
      found[cnt++] = base + bit;
      mk &= mk - 1;
    }
  }
  if (cnt == 0) found[cnt++] = n;                  // unreachable (self dist 0)
  for (int s = cnt; s < kS; ++s) found[s] = found[0];

  if (lane == 0) {
    int* out = idx + ((size_t)b * kN + n) * kS;
    #pragma unroll
    for (int s = 0; s < kS; ++s) out[s] = found[s];
  }
}

// ---------------------------------------------------------------------------
// Kernel 2: fused gather + WMMA GEMM.
// grid = (N/16, 1, B); block = 256 = 8 waves; wave w -> C_out rows [16w,16w+16).
//
// V_WMMA_F32_16X16X4_F32 fragments (ISA 7.12.2):
//   A: lanes 0-15 hold M=lane, K0/K1; lanes 16-31 hold M=lane-16, K2/K3.
//   B: lanes 0-15 hold N=lane, K0/K1; lanes 16-31 hold N=lane-16, K2/K3.
//   C/D: VGPR j, lanes 0-15 -> M=j, N=lane; lanes 16-31 -> M=j+8, N=lane-16.
//
// LDS panel layout (pair-interleaved): element (k, p) lives at
//   xs[(k>>1)*32 + p*2 + (k&1)]
// so lane (half,l15) reads the aligned float2 {K=k0+2*half, +1} for column
// l15 at dword offset k0*16 + half*32 + l15*2  ->  one ds_load_b64, banks
// 0-31 for lanes 0-15 and 32-63 for lanes 16-31 (conflict-free).
// ---------------------------------------------------------------------------
__global__ __launch_bounds__(256) void pointconv_wmma_kernel(
    const float* __restrict__ x,     // [B][C_in][N]
    const float* __restrict__ W,     // [C_out][C_in][9] == [C_out][576]
    const float* __restrict__ bias,  // [C_out]
    const int*   __restrict__ idx,   // [B][N][9]
    float* __restrict__ y)           // [B][C_out][N]
{
  __shared__ float xs[kK * kTileN];          // 576 x 16 panel, 36.9 KB
  __shared__ int   sidx[kTileN * kS];        // 144 neighbor indices

  const int b  = blockIdx.z;
  const int n0 = blockIdx.x * kTileN;
  const int t  = threadIdx.x;

  // --- stage neighbor indices for the 16-point tile ---
  if (t < kTileN * kS)
    sidx[t] = idx[((size_t)b * kN + n0) * kS + t];
  __syncthreads();

  // --- gather x into pair-interleaved LDS panel ---
  const float* xb = x + (size_t)b * kCin * kN;
  for (int i = t; i < kTileN * kS * kCin; i += 256) {
    const int pair = i % (kTileN * kS);      // 0..143
    const int c    = i / (kTileN * kS);      // 0..63
    const int p    = pair / kS;
    const int s    = pair % kS;
    const int k    = c * kS + s;
    xs[(k >> 1) * 32 + p * 2 + (k & 1)] = xb[(size_t)c * kN + sidx[pair]];
  }
  __syncthreads();

  // --- WMMA main loop: 144 x V_WMMA_F32_16X16X4_F32, unrolled x16 ---
  const int wave = t >> 5;                   // 0..7 -> output row tile
  const int lane = t & 31;
  const int half = lane >> 4;                // 0: K0/K1, 1: K2/K3
  const int l15  = lane & 15;
  const int m0   = wave * 16;

  // Per-lane A base: row (m0+l15) of W, advanced by `half` float2's.
  const v2f* __restrict__ Arow2 =
      (const v2f*)(W + (size_t)(m0 + l15) * kK) + half;
  // Per-lane B base inside the LDS panel (float2 units).
  const v2f* __restrict__ Brow2 = (const v2f*)&xs[half * 32 + l15 * 2];

  v8f acc = {};
  #pragma unroll 16
  for (int k0 = 0; k0 < kK; k0 += 4) {
    const v2f a  = Arow2[k0 >> 1];           // global_load_b64, imm offset
    const v2f bf = Brow2[(k0 >> 1) * 16];    // ds_load_b64,    imm offset
    acc = __builtin_amdgcn_wmma_f32_16x16x4_f32(
        /*neg_a=*/false, a, /*neg_b=*/false, bf,
        /*c_mod=*/(short)0, acc, /*reuse_a=*/false, /*reuse_b=*/false);
  }

  // --- bias + store (C/D layout above), immediate-offset stores ---
  const int row0 = m0 + 8 * half;
  float* __restrict__ yrow = y + ((size_t)b * kCout + row0) * kN + (n0 + l15);
  const float* __restrict__ brow = bias + row0;
  #pragma unroll
  for (int j = 0; j < 8; ++j)
    yrow[(size_t)j * kN] = acc[j] + brow[j];
}

// ---------------------------------------------------------------------------
// Host launcher. Inputs (setup_inputs order): x, pcs, W, b. Output: y fp32.
// Workspace: idx buffer, B*N*9 ints = 576 KB.
// ---------------------------------------------------------------------------
extern "C" void kernel_launch(void* const* d_in, const int* in_sizes, int n_in,
                              void* d_out, int out_size, void* d_ws, size_t ws_size,
                              hipStream_t stream) {
  const float* x    = (const float*)d_in[0];
  const float* pcs  = (const float*)d_in[1];
  const float* W    = (const float*)d_in[2];
  const float* bias = (const float*)d_in[3];
  float*       y    = (float*)d_out;
  int*         idx  = (int*)d_ws;            // B*N*9 int32

  // 1 wave per point: B*N = 16384 waves, 8 waves/block -> 2048 blocks.
  ballquery_kernel<<<dim3((kB * kN) / 8), dim3(256), 0, stream>>>(pcs, idx);

  // 16-point tiles: grid (256, 1, 4), 8 waves cover all 128 output channels.
  pointconv_wmma_kernel<<<dim3(kN / kTileN, 1, kB), dim3(256), 0, stream>>>(
      x, W, bias, idx, y);
}